// WCA_86260123174649
// MI455X (gfx1250) — compile-verified
//
#include <hip/hip_runtime.h>
#include <math.h>

#define BATCH 32
#define CH 64
#define HW 96
#define NPOS 144
#define NWIN 64
#define NG 2048
#define PLANE (HW*HW)
#define SLICE ((size_t)BATCH*CH*PLANE)   // 18,874,368 floats

typedef float v2f __attribute__((ext_vector_type(2)));
typedef float v8f __attribute__((ext_vector_type(8)));

__device__ __forceinline__ v8f wmma4(v2f a, v2f b, v8f c) {
  // D = A(16x4,f32) * B(4x16,f32) + C(16x16,f32)
  return __builtin_amdgcn_wmma_f32_16x16x4_f32(false, a, false, b, (short)0, c,
                                               false, false);
}

// ---------------------------------------------------------------------------
// 1) x_uncertain = x - (thr - |sigmoid(resize(map_s)) - thr|)
// ---------------------------------------------------------------------------
__global__ __launch_bounds__(256) void k_uncertain(
    const float* __restrict__ x, const float* __restrict__ ms,
    const float* __restrict__ thr, float* __restrict__ xu) {
  int idx = blockIdx.x * 256 + threadIdx.x;
  if (idx >= BATCH * PLANE) return;
  int b = idx / PLANE, pp = idx % PLANE;
  int y = pp / HW, xx = pp % HW;
  float sy = (y + 0.5f) * 0.5f - 0.5f;
  float sx = (xx + 0.5f) * 0.5f - 0.5f;
  int y0 = (int)floorf(sy), x0 = (int)floorf(sx);
  float fy = sy - (float)y0, fx = sx - (float)x0;
  int y0c = min(max(y0, 0), 47), y1c = min(max(y0 + 1, 0), 47);
  int x0c = min(max(x0, 0), 47), x1c = min(max(x0 + 1, 0), 47);
  const float* mp = ms + (size_t)b * 48 * 48;
  float v00 = mp[y0c * 48 + x0c], v01 = mp[y0c * 48 + x1c];
  float v10 = mp[y1c * 48 + x0c], v11 = mp[y1c * 48 + x1c];
  float mv = v00 * (1.f - fy) * (1.f - fx) + v01 * (1.f - fy) * fx +
             v10 * fy * (1.f - fx) + v11 * fy * fx;
  float s = 1.f / (1.f + __expf(-mv));
  float t = thr[0];
  float cg = t - fabsf(s - t);
  for (int c = 0; c < CH; ++c) {
    size_t o = (size_t)(b * CH + c) * PLANE + pp;
    xu[o] = x[o] - cg;
  }
}

// ---------------------------------------------------------------------------
// 2) Window conv as GEMM: out[g][o][m] = relu(sum_pk W[o][pk]*B[pk][m] + b[o])
//    "channels" = 64 window positions, columns = 144 within-window positions.
// ---------------------------------------------------------------------------
template <int TAPS>
__device__ __forceinline__ float im2col_win(const float* s_img, int pk, int mi,
                                            int mj) {
  int p, ii, jj;
  if (TAPS == 1) {
    p = pk; ii = mi; jj = mj;
  } else {
    p = pk / 9;
    int t = pk - p * 9;
    ii = mi + t / 3 - 1;
    jj = mj + t % 3 - 1;
  }
  if ((unsigned)ii >= 12u || (unsigned)jj >= 12u) return 0.f;  // window pad
  return s_img[((p >> 3) * 12 + ii) * HW + (p & 7) * 12 + jj];
}

template <int TAPS>
__global__ __launch_bounds__(256) void k_winconv(
    const float* __restrict__ src, const float* __restrict__ w,
    const float* __restrict__ bias, float* __restrict__ out) {
  __shared__ float s_img[PLANE];  // 36 KB: one (b,c) plane
  int g = blockIdx.x;
  const float* sp = src + (size_t)g * PLANE;
  for (int i = threadIdx.x; i < PLANE; i += 256) s_img[i] = sp[i];
  __syncthreads();
  const int K = NWIN * TAPS;
  int wave = threadIdx.x >> 5, lane = threadIdx.x & 31;
  int lrow = lane & 15, lhi = lane >> 4;
  float* op = out + (size_t)g * CH * NPOS;
  for (int tile = wave; tile < 4 * 9; tile += 8) {
    int o0 = (tile / 9) * 16, n0 = (tile % 9) * 16;
    int m = n0 + lrow;
    int mi = m / 12, mj = m % 12;
    v8f acc = {0, 0, 0, 0, 0, 0, 0, 0};
    for (int k0 = 0; k0 < K; k0 += 4) {
      int ka = k0 + 2 * lhi;
      v2f a, bf;
      a.x = w[(o0 + lrow) * K + ka];
      a.y = w[(o0 + lrow) * K + ka + 1];
      bf.x = im2col_win<TAPS>(s_img, ka, mi, mj);
      bf.y = im2col_win<TAPS>(s_img, ka + 1, mi, mj);
      acc = wmma4(a, bf, acc);
    }
#pragma unroll
    for (int vv = 0; vv < 8; ++vv) {
      int o = o0 + vv + 8 * lhi;
      op[o * NPOS + m] = fmaxf(acc[vv] + bias[o], 0.f);
    }
  }
}

// ---------------------------------------------------------------------------
// 3) Attention per g: S = scale*q^T k + bias; softmax; O = v * S^T,
//    scattered directly into image layout (fused window_reverse).
// ---------------------------------------------------------------------------
__global__ __launch_bounds__(256) void k_attn(
    const float* __restrict__ q, const float* __restrict__ k,
    const float* __restrict__ v, const float* __restrict__ rpb,
    float* __restrict__ xr) {
  __shared__ float s_k[CH * NPOS];       // 36 KB
  __shared__ float s_v[CH * NPOS];       // 36 KB
  __shared__ float s_S[8][16 * NPOS];    // 72 KB: per-wave 16-row score slab
  __shared__ float s_rpb[529];
  int g = blockIdx.x;
  const float* qg = q + (size_t)g * CH * NPOS;
  const float* kg = k + (size_t)g * CH * NPOS;
  const float* vg = v + (size_t)g * CH * NPOS;
  for (int i = threadIdx.x; i < CH * NPOS; i += 256) {
    s_k[i] = kg[i];
    s_v[i] = vg[i];
  }
  for (int i = threadIdx.x; i < 529; i += 256) s_rpb[i] = rpb[i];
  __syncthreads();
  int wave = threadIdx.x >> 5, lane = threadIdx.x & 31;
  int lrow = lane & 15, lhi = lane >> 4;
  float* Sw = s_S[wave];
  const float scale = 0.125f;  // 64^-0.5
  for (int rb = wave; rb < 9; rb += 8) {
    int m10 = rb * 16;
    // S row-block (16 x 144)
    for (int nt = 0; nt < 9; ++nt) {
      int n0 = nt * 16;
      v8f acc = {0, 0, 0, 0, 0, 0, 0, 0};
      for (int k0 = 0; k0 < CH; k0 += 4) {
        int ka = k0 + 2 * lhi;
        v2f a, bf;
        a.x = qg[ka * NPOS + m10 + lrow];        // A = q^T
        a.y = qg[(ka + 1) * NPOS + m10 + lrow];
        bf.x = s_k[ka * NPOS + n0 + lrow];       // B = k
        bf.y = s_k[(ka + 1) * NPOS + n0 + lrow];
        acc = wmma4(a, bf, acc);
      }
      int m2 = n0 + lrow;
      int i2 = m2 / 12, j2 = m2 % 12;
#pragma unroll
      for (int vv = 0; vv < 8; ++vv) {
        int m1 = m10 + vv + 8 * lhi;
        int i1 = m1 / 12, j1 = m1 % 12;
        float bia = s_rpb[(i1 - i2 + 11) * 23 + (j1 - j2 + 11)];
        Sw[(vv + 8 * lhi) * NPOS + m2] = acc[vv] * scale + bia;
      }
    }
    // softmax: lanes 0..15 each own one row (same-wave LDS, in-order)
    if (lane < 16) {
      float* row = &Sw[lane * NPOS];
      float mx = -1e30f;
      for (int c2 = 0; c2 < NPOS; ++c2) mx = fmaxf(mx, row[c2]);
      float sum = 0.f;
      for (int c2 = 0; c2 < NPOS; ++c2) {
        float e = __expf(row[c2] - mx);
        row[c2] = e;
        sum += e;
      }
      float inv = 1.f / sum;
      for (int c2 = 0; c2 < NPOS; ++c2) row[c2] *= inv;
    }
    // O = v (64x144) * attn^T (144x16); scatter with window_reverse
    for (int mt = 0; mt < 4; ++mt) {
      int o0 = mt * 16;
      v8f acc = {0, 0, 0, 0, 0, 0, 0, 0};
      for (int k0 = 0; k0 < NPOS; k0 += 4) {
        int ka = k0 + 2 * lhi;
        v2f a, bf;
        a.x = s_v[(o0 + lrow) * NPOS + ka];
        a.y = s_v[(o0 + lrow) * NPOS + ka + 1];
        bf.x = Sw[lrow * NPOS + ka];
        bf.y = Sw[lrow * NPOS + ka + 1];
        acc = wmma4(a, bf, acc);
      }
      int m1 = m10 + lrow;
      int ii = m1 / 12, jj = m1 % 12;
#pragma unroll
      for (int vv = 0; vv < 8; ++vv) {
        int o = o0 + vv + 8 * lhi;  // o == window position (hb,wb)
        xr[(size_t)g * PLANE + ((o >> 3) * 12 + ii) * HW + (o & 7) * 12 + jj] =
            acc[vv];
      }
    }
  }
}

// ---------------------------------------------------------------------------
// 4) Image-space 3x3 conv (SAME, image-edge zero pad) as im2col WMMA GEMM.
//    CIN2==2 reads concat(in0, in1) -> fused concat for o2.
// ---------------------------------------------------------------------------
template <int CIN2>
__global__ __launch_bounds__(256) void k_conv3(
    const float* __restrict__ in0, const float* __restrict__ in1,
    const float* __restrict__ w, const float* __restrict__ bias,
    float* __restrict__ out, int relu) {
  constexpr int CIN = 64 * CIN2;
  constexpr int TH = (CIN2 == 1) ? 16 : 8;  // tile height (x16 wide)
  constexpr int PR = TH + 2;
  constexpr int PATCH = PR * 18;
  __shared__ float s_in[CIN * PATCH];  // 81 KB (64ch) / 90 KB (128ch)
  constexpr int TB = (HW / TH) * 6;    // tiles per image
  int b = blockIdx.x / TB;
  int t = blockIdx.x % TB;
  int ty = (t / 6) * TH, tx = (t % 6) * 16;
  for (int idx = threadIdx.x; idx < CIN * PATCH; idx += 256) {
    int cin = idx / PATCH, pp = idx % PATCH;
    int py = pp / 18, px = pp % 18;
    int gy = ty + py - 1, gx = tx + px - 1;
    float val = 0.f;
    if ((unsigned)gy < (unsigned)HW && (unsigned)gx < (unsigned)HW) {
      const float* src = (cin < 64)
                             ? in0 + (size_t)(b * 64 + cin) * PLANE
                             : in1 + (size_t)(b * 64 + cin - 64) * PLANE;
      val = src[gy * HW + gx];
    }
    s_in[idx] = val;
  }
  __syncthreads();
  const int K = CIN * 9;
  int wave = threadIdx.x >> 5, lane = threadIdx.x & 31;
  int lrow = lane & 15, lhi = lane >> 4;
  constexpr int NTILE = TH;  // 16*TH pixels / 16
  for (int tile = wave; tile < 4 * NTILE; tile += 8) {
    int o0 = (tile / NTILE) * 16, n0 = (tile % NTILE) * 16;
    int m = n0 + lrow;
    int py = m / 16, px = m % 16;
    v8f acc = {0, 0, 0, 0, 0, 0, 0, 0};
    for (int k0 = 0; k0 < K; k0 += 4) {
      int ka = k0 + 2 * lhi;
      v2f a, bf;
      a.x = w[(o0 + lrow) * K + ka];
      a.y = w[(o0 + lrow) * K + ka + 1];
      int c0 = ka / 9, t0 = ka - c0 * 9;
      bf.x = s_in[c0 * PATCH + (py + t0 / 3) * 18 + px + t0 % 3];
      int pk1 = ka + 1;
      int c1 = pk1 / 9, t1 = pk1 - c1 * 9;
      bf.y = s_in[c1 * PATCH + (py + t1 / 3) * 18 + px + t1 % 3];
      acc = wmma4(a, bf, acc);
    }
#pragma unroll
    for (int vv = 0; vv < 8; ++vv) {
      int o = o0 + vv + 8 * lhi;
      float r = acc[vv] + bias[o];
      if (relu) r = fmaxf(r, 0.f);
      out[(size_t)(b * 64 + o) * PLANE + (ty + py) * HW + tx + px] = r;
    }
  }
}

// ---------------------------------------------------------------------------
// 5) 1x1 conv 64 -> 1
// ---------------------------------------------------------------------------
__global__ __launch_bounds__(256) void k_conv1(
    const float* __restrict__ in, const float* __restrict__ w,
    const float* __restrict__ bias, float* __restrict__ out) {
  int idx = blockIdx.x * 256 + threadIdx.x;
  if (idx >= BATCH * PLANE) return;
  int b = idx / PLANE, pp = idx % PLANE;
  float s = bias[0];
  for (int c = 0; c < CH; ++c)
    s += in[(size_t)(b * CH + c) * PLANE + pp] * w[c];
  out[idx] = s;
}

extern "C" void kernel_launch(void* const* d_in, const int* in_sizes, int n_in,
                              void* d_out, int out_size, void* d_ws,
                              size_t ws_size, hipStream_t stream) {
  (void)in_sizes; (void)n_in; (void)out_size; (void)ws_size;
  const float* x    = (const float*)d_in[0];
  const float* mapS = (const float*)d_in[1];
  const float* thr  = (const float*)d_in[2];
  const float* rpb  = (const float*)d_in[3];
  const float* q_w  = (const float*)d_in[4];
  const float* q_b  = (const float*)d_in[5];
  const float* k_w  = (const float*)d_in[6];
  const float* k_b  = (const float*)d_in[7];
  const float* v_w  = (const float*)d_in[8];
  const float* v_b  = (const float*)d_in[9];
  const float* o1_w = (const float*)d_in[10];
  const float* o1_b = (const float*)d_in[11];
  const float* o2_w = (const float*)d_in[12];
  const float* o2_b = (const float*)d_in[13];
  const float* o3_w = (const float*)d_in[14];
  const float* o3_b = (const float*)d_in[15];
  const float* o4_w = (const float*)d_in[16];
  const float* o4_b = (const float*)d_in[17];

  float* buf = (float*)d_ws;
  float* xu  = buf;              // later reused as o2 output
  float* qb_ = buf + SLICE;      // later reused as o1 output
  float* kb_ = buf + 2 * SLICE;
  float* vb_ = buf + 3 * SLICE;
  float* xrb = buf + 4 * SLICE;
  float* xf  = (float*)d_out;            // xf after o3 (first output)
  float* op  = xf + SLICE;               // final 1-channel map (second output)

  dim3 blk(256);
  k_uncertain<<<(BATCH * PLANE + 255) / 256, blk, 0, stream>>>(x, mapS, thr, xu);
  k_winconv<9><<<NG, blk, 0, stream>>>(x, q_w, q_b, qb_);
  k_winconv<1><<<NG, blk, 0, stream>>>(xu, k_w, k_b, kb_);
  k_winconv<1><<<NG, blk, 0, stream>>>(xu, v_w, v_b, vb_);
  k_attn<<<NG, blk, 0, stream>>>(qb_, kb_, vb_, rpb, xrb);
  k_conv3<1><<<32 * 6 * 6, blk, 0, stream>>>(xrb, nullptr, o1_w, o1_b, qb_, 1);
  k_conv3<2><<<32 * 12 * 6, blk, 0, stream>>>(x, qb_, o2_w, o2_b, xu, 1);
  k_conv3<1><<<32 * 6 * 6, blk, 0, stream>>>(xu, nullptr, o3_w, o3_b, xf, 1);
  k_conv1<<<(BATCH * PLANE + 255) / 256, blk, 0, stream>>>(xf, o4_w, o4_b, op);
}